// QuanvLayer1D_39642548142545
// MI455X (gfx1250) — compile-verified
//
#include <hip/hip_runtime.h>
#include <hip/hip_bf16.h>

#define BS        16
#define CH        8
#define LEN       4096
#define L_OUT     4097          // (4096 + 2*1 - 2) + 1
#define NTILE     257           // ceil(4097/16) position tiles per batch
#define FEATURES  10

typedef __attribute__((ext_vector_type(2))) float v2f;
typedef __attribute__((ext_vector_type(8))) float v8f;

// ---------------- workspace layout (floats) ----------------
// [0..15]      s0 real  (state after basic_layer(weights[0]) applied to |0000>)
// [16..31]     s0 imag
// [32..287]    U1 real  (row-major 16x16: [out_state][in_state])
// [288..543]   U1 imag
// [544..799]   U2 real
// [800..1055]  U2 imag
// [1056..1119] maxv[b][4]  (nonneg floats; atomicMax via uint monotonicity)
#define WS_S0R  0
#define WS_S0I  16
#define WS_U1R  32
#define WS_U1I  288
#define WS_U2R  544
#define WS_U2I  800
#define WS_MAXV 1056

// ============ precompute: simulate basic_layer on basis states ============
// Qubit j is bit (3-j) of the flat index (C-order flatten of (2,2,2,2)).

template<int M>
__device__ inline void rx_gate(float* re, float* im, float theta) {
  float s, c;
  sincosf(0.5f * theta, &s, &c);
  // RX = [[c, -i s], [-i s, c]]
#pragma unroll
  for (int i = 0; i < 16; ++i) {
    if ((i & M) == 0) {
      const int j = i | M;
      float ar = re[i], ai = im[i], br = re[j], bi = im[j];
      re[i] =  c * ar + s * bi;  im[i] =  c * ai - s * br;
      re[j] =  s * ai + c * br;  im[j] = -s * ar + c * bi;
    }
  }
}

template<int MC, int MT>
__device__ inline void cnot_gate(float* re, float* im) {
#pragma unroll
  for (int i = 0; i < 16; ++i) {
    if ((i & MC) && !(i & MT)) {
      const int j = i | MT;
      float t;
      t = re[i]; re[i] = re[j]; re[j] = t;
      t = im[i]; im[i] = im[j]; im[j] = t;
    }
  }
}

__device__ inline void basic_layer(const float* w, float* re, float* im) {
  rx_gate<8>(re, im, w[0]);   // qubit 0
  rx_gate<4>(re, im, w[1]);   // qubit 1
  rx_gate<2>(re, im, w[2]);   // qubit 2
  rx_gate<1>(re, im, w[3]);   // qubit 3
  cnot_gate<8, 4>(re, im);    // CNOT 0->1
  cnot_gate<4, 2>(re, im);    // CNOT 1->2
  cnot_gate<2, 1>(re, im);    // CNOT 2->3
  cnot_gate<1, 8>(re, im);    // CNOT 3->0
}

__global__ void precompute_kernel(const float* __restrict__ weights,
                                  float* __restrict__ ws) {
  const int t = threadIdx.x;
  if (t >= 16) return;
#pragma unroll
  for (int l = 0; l < 3; ++l) {
    float re[16], im[16];
#pragma unroll
    for (int i = 0; i < 16; ++i) { re[i] = 0.f; im[i] = 0.f; }
    re[t] = 1.f;                       // basis column t
    basic_layer(weights + l * 4, re, im);
    if (l == 0) {
      if (t == 0) {                    // s0 = U0 e0
#pragma unroll
        for (int i = 0; i < 16; ++i) { ws[WS_S0R + i] = re[i]; ws[WS_S0I + i] = im[i]; }
      }
    } else {
      float* Ur = ws + (l == 1 ? WS_U1R : WS_U2R);
      float* Ui = ws + (l == 1 ? WS_U1I : WS_U2I);
#pragma unroll
      for (int i = 0; i < 16; ++i) { Ur[i * 16 + t] = re[i]; Ui[i * 16 + t] = im[i]; }
    }
  }
}

// ============ main kernel helpers ============

// D-layout (WMMA 16x16 f32 C/D): lane L holds column N = L&15;
// lanes 0-15 hold rows 0..7 in regs d[0..7], lanes 16-31 hold rows 8..15.

__device__ inline void ry2(float& a, float& b, float c, float s) {
  float x = a, y = b;
  a = c * x - s * y;
  b = s * x + c * y;
}

// Encoding step: RY(w0) on qubits 0,2 and RY(w1) on qubits 1,3.
__device__ inline void ry_encode(float* dr, float* di, bool lo,
                                 float c0, float s0, float c1, float s1) {
  // qubit 0 (angle w0): rows (v, v+8) — cross lane-half pair
#pragma unroll
  for (int v = 0; v < 8; ++v) {
    float orr = __shfl_xor(dr[v], 16);
    float oii = __shfl_xor(di[v], 16);
    dr[v] = lo ? (c0 * dr[v] - s0 * orr) : (c0 * dr[v] + s0 * orr);
    di[v] = lo ? (c0 * di[v] - s0 * oii) : (c0 * di[v] + s0 * oii);
  }
  // qubit 1 (angle w1): rows (v, v+4) — within lane
#pragma unroll
  for (int v = 0; v < 4; ++v) {
    ry2(dr[v], dr[v + 4], c1, s1);
    ry2(di[v], di[v + 4], c1, s1);
  }
  // qubit 2 (angle w0): rows (v, v+2) for v in {0,1,4,5}
#pragma unroll
  for (int g = 0; g < 2; ++g) {
#pragma unroll
    for (int v = 0; v < 2; ++v) {
      int a = g * 4 + v;
      ry2(dr[a], dr[a + 2], c0, s0);
      ry2(di[a], di[a + 2], c0, s0);
    }
  }
  // qubit 3 (angle w1): rows (v, v+1) for v in {0,2,4,6}
#pragma unroll
  for (int v = 0; v < 8; v += 2) {
    ry2(dr[v], dr[v + 1], c1, s1);
    ry2(di[v], di[v + 1], c1, s1);
  }
}

// Convert D-layout state to WMMA B-layout (4x16 per K-block kb):
// B[kb] vgpr0 = S[4kb + (lo?0:2)][col], vgpr1 = S[4kb + (lo?1:3)][col]
__device__ inline void d_to_b(const float* d, bool lo, v2f* B) {
  float o[8];
#pragma unroll
  for (int v = 0; v < 8; ++v) o[v] = __shfl_xor(d[v], 16);
  B[0][0] = lo ? d[0] : o[2];  B[0][1] = lo ? d[1] : o[3];
  B[1][0] = lo ? d[4] : o[6];  B[1][1] = lo ? d[5] : o[7];
  B[2][0] = lo ? o[0] : d[2];  B[2][1] = lo ? o[1] : d[3];
  B[3][0] = lo ? o[4] : d[6];  B[3][1] = lo ? o[5] : d[7];
}

__device__ inline v8f wmma4(v2f a, v2f b, v8f c) {
  return __builtin_amdgcn_wmma_f32_16x16x4_f32(false, a, false, b, (short)0, c,
                                               false, false);
}

// Complex 16x16 matmul: D = U * S   (Dr = UrSr - UiSi, Di = UrSi + UiSr)
__device__ inline void cmatmul(const v2f* Ar, const v2f* Ai,
                               const v2f* Br, const v2f* Bi,
                               float* dr, float* di) {
  v8f cr, ci;
#pragma unroll
  for (int v = 0; v < 8; ++v) { cr[v] = 0.f; ci[v] = 0.f; }
#pragma unroll
  for (int kb = 0; kb < 4; ++kb) {
    v2f nbi; nbi[0] = -Bi[kb][0]; nbi[1] = -Bi[kb][1];  // F32 WMMA has no A/B neg
    cr = wmma4(Ar[kb], Br[kb], cr);
    cr = wmma4(Ai[kb], nbi,    cr);
    ci = wmma4(Ar[kb], Bi[kb], ci);
    ci = wmma4(Ai[kb], Br[kb], ci);
  }
#pragma unroll
  for (int v = 0; v < 8; ++v) { dr[v] = cr[v]; di[v] = ci[v]; }
}

// ============ main kernel: one wave = 16 positions of one batch, all channels ============
__global__ __launch_bounds__(256) void quanv_main_kernel(
    const float* __restrict__ x, const float* __restrict__ ws,
    float* __restrict__ maxv) {
  const int lane = threadIdx.x & 31;
  const bool lo  = lane < 16;
  const int col  = lane & 15;
  const int gw   = blockIdx.x * (blockDim.x >> 5) + (threadIdx.x >> 5);
  const int b    = gw / NTILE;
  const int tile = gw - b * NTILE;
  const int p    = tile * 16 + col;   // output position handled by this lane's column

  // Load U1/U2 in WMMA A-layout: lane L: M = L&15; vgpr0/1 = K = 4kb+(lo?0:2)+{0,1}
  v2f a1r[4], a1i[4], a2r[4], a2i[4];
  {
    const int k0 = lo ? 0 : 2;
#pragma unroll
    for (int kb = 0; kb < 4; ++kb) {
      const int idx = col * 16 + kb * 4 + k0;
      a1r[kb][0] = ws[WS_U1R + idx]; a1r[kb][1] = ws[WS_U1R + idx + 1];
      a1i[kb][0] = ws[WS_U1I + idx]; a1i[kb][1] = ws[WS_U1I + idx + 1];
      a2r[kb][0] = ws[WS_U2R + idx]; a2r[kb][1] = ws[WS_U2R + idx + 1];
      a2i[kb][0] = ws[WS_U2I + idx]; a2i[kb][1] = ws[WS_U2I + idx + 1];
    }
  }
  // s0 replicated into D-layout (same for every column)
  const int rbase = lo ? 0 : 8;
  float s0r[8], s0i[8];
#pragma unroll
  for (int v = 0; v < 8; ++v) {
    s0r[v] = ws[WS_S0R + rbase + v];
    s0i[v] = ws[WS_S0I + rbase + v];
  }

  float ev_acc[4] = {0.f, 0.f, 0.f, 0.f};
  const float* xb = x + (size_t)b * CH * LEN;

  for (int ch = 0; ch < CH; ++ch) {
    const float* xc = xb + (size_t)ch * LEN;
    // window at padded position p: (xp[p], xp[p+1]); PADDING=1
    float w0 = (p >= 1 && p <= LEN) ? xc[p - 1] : 0.f;
    float w1 = (p < LEN) ? xc[p] : 0.f;
    float c0, s0a, c1, s1;
    sincosf(0.5f * w0, &s0a, &c0);
    sincosf(0.5f * w1, &s1, &c1);

    float dr[8], di[8];
#pragma unroll
    for (int v = 0; v < 8; ++v) { dr[v] = s0r[v]; di[v] = s0i[v]; }

    ry_encode(dr, di, lo, c0, s0a, c1, s1);
    v2f br[4], bi[4];
    d_to_b(dr, lo, br);
    d_to_b(di, lo, bi);
    cmatmul(a1r, a1i, br, bi, dr, di);     // 16 WMMAs

    ry_encode(dr, di, lo, c0, s0a, c1, s1);
    d_to_b(dr, lo, br);
    d_to_b(di, lo, bi);
    cmatmul(a2r, a2i, br, bi, dr, di);     // 16 WMMAs

    // Pauli-Z expectations: ev[j] = sum_i |amp_i|^2 * (1 - 2*bit_{3-j}(i))
    float ev[4] = {0.f, 0.f, 0.f, 0.f};
#pragma unroll
    for (int v = 0; v < 8; ++v) {
      const int row = rbase + v;
      float p2 = dr[v] * dr[v] + di[v] * di[v];
#pragma unroll
      for (int j = 0; j < 4; ++j)
        ev[j] += ((row >> (3 - j)) & 1) ? -p2 : p2;
    }
#pragma unroll
    for (int j = 0; j < 4; ++j) {
      ev[j] += __shfl_xor(ev[j], 16);      // combine row halves
      ev_acc[j] += ev[j];                  // channel sum
    }
  }

  // relu + max over the 16 positions this wave owns, then global atomicMax.
  const bool valid = (p < L_OUT);
#pragma unroll
  for (int j = 0; j < 4; ++j) {
    float mj = valid ? fmaxf(ev_acc[j], 0.f) : 0.f;   // relu; 0 is the relu floor
    mj = fmaxf(mj, __shfl_xor(mj, 1));
    mj = fmaxf(mj, __shfl_xor(mj, 2));
    mj = fmaxf(mj, __shfl_xor(mj, 4));
    mj = fmaxf(mj, __shfl_xor(mj, 8));
    if (lane == 0)   // nonneg floats are monotone under uint compare
      atomicMax((unsigned int*)&maxv[b * 4 + j], __float_as_uint(mj));
  }
}

// ============ epilogue kernels ============
__global__ void init_kernel(float* __restrict__ maxv) {
  const int t = threadIdx.x;
  if (t < BS * 4) maxv[t] = 0.f;
}

__global__ void dense_kernel(const float* __restrict__ maxv,
                             const float* __restrict__ dw,
                             const float* __restrict__ db,
                             float* __restrict__ out) {
  const int t = threadIdx.x;
  if (t >= BS * FEATURES) return;
  const int b = t / FEATURES, f = t - b * FEATURES;
  float acc = db[f];
#pragma unroll
  for (int j = 0; j < 4; ++j)
    acc += maxv[b * 4 + j] * dw[j * FEATURES + f];
  out[t] = acc;
}

extern "C" void kernel_launch(void* const* d_in, const int* in_sizes, int n_in,
                              void* d_out, int out_size, void* d_ws, size_t ws_size,
                              hipStream_t stream) {
  const float* x  = (const float*)d_in[0];   // (16, 8, 4096)
  const float* w  = (const float*)d_in[1];   // (3, 4)
  const float* dw = (const float*)d_in[2];   // (4, 10)
  const float* db = (const float*)d_in[3];   // (10,)
  float* ws   = (float*)d_ws;
  float* out  = (float*)d_out;               // (16, 10)
  float* maxv = ws + WS_MAXV;

  init_kernel<<<1, 64, 0, stream>>>(maxv);
  precompute_kernel<<<1, 32, 0, stream>>>(w, ws);
  const int total_waves = BS * NTILE;        // 4112 (exact multiple of 8)
  quanv_main_kernel<<<total_waves / 8, 256, 0, stream>>>(x, ws, maxv);
  dense_kernel<<<1, 256, 0, stream>>>(maxv, dw, db, out);
}